// OrthoLoss_27255862460737
// MI455X (gfx1250) — compile-verified
//
#include <hip/hip_runtime.h>

typedef __attribute__((ext_vector_type(2))) float v2f;
typedef __attribute__((ext_vector_type(8))) float v8f;

#define BSZ 256   // batch (rows of embeddings)
#define DIM 1024  // embedding dim

// ---------------------------------------------------------------------------
// Kernel 1: G = E @ E^T via V_WMMA_F32_16X16X4_F32 (full fp32 precision).
// One wave (32 lanes) computes one 16x16 tile of G. K-loop steps by 4.
// A-tile layout (16x4 f32): lane L -> row i0+(L&15), VGPR r -> K = 2*(L>>4)+r
// B-tile layout (4x16 f32): lane L -> col j0+(L&15), VGPR r -> K = 2*(L>>4)+r
// (B = E^T tile, so both sides load E identically with different row bases.)
// C/D layout: VGPR r -> M = r + 8*(L>>4), N = L&15.
// ---------------------------------------------------------------------------
__global__ __launch_bounds__(32) void gram_wmma_f32(const float* __restrict__ E,
                                                    float* __restrict__ G) {
    const int lane = threadIdx.x;
    const int half = lane >> 4;   // 0 or 1
    const int m    = lane & 15;
    const int i0   = blockIdx.y * 16;
    const int j0   = blockIdx.x * 16;

    const float* pa = E + (size_t)(i0 + m) * DIM + half * 2;
    const float* pb = E + (size_t)(j0 + m) * DIM + half * 2;

    v8f c = {};
    #pragma unroll 4
    for (int k = 0; k < DIM; k += 4) {
        v2f a, b;
        a.x = pa[k];     a.y = pa[k + 1];
        b.x = pb[k];     b.y = pb[k + 1];
        // 8 args: (neg_a, A, neg_b, B, c_mod, C, reuse_a, reuse_b)
        c = __builtin_amdgcn_wmma_f32_16x16x4_f32(false, a, false, b,
                                                  (short)0, c, false, false);
    }

    #pragma unroll
    for (int r = 0; r < 8; ++r) {
        const int M = r + half * 8;
        G[(size_t)(i0 + M) * BSZ + (j0 + m)] = c[r];
    }
}

// ---------------------------------------------------------------------------
// Kernel 2: dist[i,j] = sqrt(max(G[i,i] - 2 G[i,j] + G[j,j], 0))
// ---------------------------------------------------------------------------
__global__ __launch_bounds__(BSZ) void dist_kernel(const float* __restrict__ G,
                                                   float* __restrict__ dist) {
    const int i = blockIdx.x;
    const int j = threadIdx.x;
    const float gii = G[(size_t)i * BSZ + i];
    const float gjj = G[(size_t)j * BSZ + j];
    const float gij = G[(size_t)i * BSZ + j];
    const float d2  = fmaxf(gii - 2.0f * gij + gjj, 0.0f);
    dist[(size_t)i * BSZ + j] = sqrtf(d2);
}

// ---------------------------------------------------------------------------
// Per-triplet loss (exact reference arithmetic, fp32).
// dot = g[i] - G[i,k] - G[i,j] + G[j,k]; ap = dist[i,j]; an = dist[i,k]
// ---------------------------------------------------------------------------
__device__ __forceinline__ float triplet_L(const float* __restrict__ G,
                                           const float* __restrict__ dist,
                                           int i, int j, int k) {
    const float gi   = G[(size_t)i * BSZ + i];
    const float Gij  = G[(size_t)i * BSZ + j];
    const float Gik  = G[(size_t)i * BSZ + k];
    const float Gjk  = G[(size_t)j * BSZ + k];
    const float ap   = dist[(size_t)i * BSZ + j];
    const float an   = dist[(size_t)i * BSZ + k];
    const float dot  = gi - Gik - Gij + Gjk;
    const float ct   = dot / (an * ap);
    const float s    = sqrtf(fmaxf(1.0f - ct * ct, 0.0f));
    const float anct = an * ct;
    float t1 = 0.5f * ap - anct;  t1 *= t1;
    float t2 = (ap - anct) * ct / s;  t2 *= t2;
    return t1 + t2;
}

// ---------------------------------------------------------------------------
// Kernel 3: sum over all valid triplets. Block = one (i,j) pair, thread = k.
// mask: i!=j, i!=k, j!=k, y[i]==y[j], y[i]!=y[k]
// ---------------------------------------------------------------------------
__global__ __launch_bounds__(BSZ) void reduce_kernel(const float* __restrict__ G,
                                                     const float* __restrict__ dist,
                                                     const int* __restrict__ y,
                                                     float* __restrict__ out) {
    const int i = blockIdx.x >> 8;
    const int j = blockIdx.x & 255;
    const int yi = y[i];
    if (i == j || y[j] != yi) return;   // uniform early exit, no barriers yet

    const int k = threadIdx.x;
    float v = 0.0f;
    if (k != i && k != j && y[k] != yi) {
        v = triplet_L(G, dist, i, j, k);
    }

    __shared__ float red[BSZ];
    red[k] = v;
    __syncthreads();
    #pragma unroll
    for (int off = BSZ / 2; off > 0; off >>= 1) {
        if (k < off) red[k] += red[k + off];
        __syncthreads();
    }
    if (k == 0) atomicAdd(out, red[0]);
}

// ---------------------------------------------------------------------------
// Kernel 4: add L at the first masked flat index (row-major i,j,k), matching
// the reference's `total + L.reshape(-1)[argmax(mask)]`. If no mask exists,
// nothing is added (sum is already 0 -> matches where(mask.any(), ..., 0)).
// ---------------------------------------------------------------------------
__global__ void first_term_kernel(const float* __restrict__ G,
                                  const float* __restrict__ dist,
                                  const int* __restrict__ y,
                                  float* __restrict__ out) {
    if (threadIdx.x != 0 || blockIdx.x != 0) return;
    for (int i = 0; i < BSZ; ++i) {
        const int yi = y[i];
        for (int j = 0; j < BSZ; ++j) {
            if (j == i || y[j] != yi) continue;
            for (int k = 0; k < BSZ; ++k) {
                if (k == i || k == j || y[k] == yi) continue;
                atomicAdd(out, triplet_L(G, dist, i, j, k));
                return;
            }
        }
    }
}

extern "C" void kernel_launch(void* const* d_in, const int* in_sizes, int n_in,
                              void* d_out, int out_size, void* d_ws, size_t ws_size,
                              hipStream_t stream) {
    const float* E = (const float*)d_in[0];   // (256, 1024) fp32
    const int*   y = (const int*)d_in[1];     // (256,) labels
    float* out  = (float*)d_out;              // scalar fp32
    float* G    = (float*)d_ws;               // 256*256 floats
    float* dist = G + BSZ * BSZ;              // 256*256 floats

    hipMemsetAsync(out, 0, sizeof(float), stream);

    gram_wmma_f32<<<dim3(16, 16), 32, 0, stream>>>(E, G);
    dist_kernel<<<BSZ, BSZ, 0, stream>>>(G, dist);
    reduce_kernel<<<BSZ * BSZ, BSZ, 0, stream>>>(G, dist, y, out);
    first_term_kernel<<<1, 32, 0, stream>>>(G, dist, y, out);
}